// WindowLayer_32950989095055
// MI455X (gfx1250) — compile-verified
//
#include <hip/hip_runtime.h>

// Problem constants (from reference)
#define Bn 1024
#define Dn 1024
#define Kn 20
#define Sn 1024
#define Ln 128

typedef float v2f __attribute__((ext_vector_type(2)));
typedef float v8f __attribute__((ext_vector_type(8)));

// ---------------------------------------------------------------------------
// Kernel A: fused projection GEMM via V_WMMA_F32_16X16X4_F32.
// grid = (M tiles = 64, N tiles = 2, matrix = 3 {Wa,Wb,Wk}), block = 1 wave32.
// Each wave computes a 16x16 tile of x @ W, then applies bias + exp
// (+ prev_kappa for the kappa matrix) and scatters to workspace / output.
// ---------------------------------------------------------------------------
__global__ __launch_bounds__(32)
void proj_wmma_kernel(const float* __restrict__ x,
                      const float* __restrict__ Wa, const float* __restrict__ Wb,
                      const float* __restrict__ Wk,
                      const float* __restrict__ ba, const float* __restrict__ bb,
                      const float* __restrict__ bk,
                      const float* __restrict__ prev_kappa,
                      float* __restrict__ alpha_ws, float* __restrict__ beta_ws,
                      float* __restrict__ kappa_ws, float* __restrict__ kappa_out)
{
    const int mtile = blockIdx.x;   // 0..63  -> rows  mtile*16 .. +15
    const int ntile = blockIdx.y;   // 0..1   -> cols  ntile*16 .. +15 (20 valid)
    const int mat   = blockIdx.z;   // 0=alpha, 1=beta, 2=kappa

    const int lane = threadIdx.x;   // 0..31 (wave32)
    const int half = lane >> 4;     // 0: K-pair {0,1}; 1: K-pair {2,3}
    const int lm   = lane & 15;

    const float* W    = (mat == 0) ? Wa : (mat == 1) ? Wb : Wk;
    const float* bias = (mat == 0) ? ba : (mat == 1) ? bb : bk;

    // A-matrix (16x4 f32): lane lm = row M, regs hold K = 2*half + {0,1}
    const int   mrow = mtile * 16 + lm;
    // B-matrix (4x16 f32): lane lm = col N, regs hold K = 2*half + {0,1}
    const int   n     = ntile * 16 + lm;
    const int   nc    = (n < Kn) ? n : (Kn - 1);      // clamp: loads in bounds
    const float nmask = (n < Kn) ? 1.0f : 0.0f;       // zero padded columns

    const float* xrow = x + (size_t)mrow * Dn + 2 * half;       // contiguous pair
    const float* wcol = W + (size_t)(2 * half) * Kn + nc;       // stride Kn rows

    v8f acc = {0.f, 0.f, 0.f, 0.f, 0.f, 0.f, 0.f, 0.f};

    for (int kk = 0; kk < Dn; kk += 4) {
        v2f a = *reinterpret_cast<const v2f*>(xrow + kk);       // b64 load
        v2f b;
        b.x = nmask * wcol[(size_t)kk * Kn];
        b.y = nmask * wcol[(size_t)kk * Kn + Kn];
        // D = A(16x4) * B(4x16) + C ; 8 args: neg_a, A, neg_b, B, c_mod, C, reuse_a, reuse_b
        acc = __builtin_amdgcn_wmma_f32_16x16x4_f32(false, a, false, b,
                                                    (short)0, acc, false, false);
    }

    // C/D layout: lane lm = col N; VGPR j holds row M = 8*half + j
    const int ncol = ntile * 16 + lm;
    if (ncol < Kn) {
        const float bval = bias[ncol];
        #pragma unroll
        for (int j = 0; j < 8; ++j) {
            const int    row = mtile * 16 + 8 * half + j;
            const float  e   = __expf(acc[j] + bval);
            const size_t idx = (size_t)row * Kn + ncol;
            if (mat == 0) {
                alpha_ws[idx] = e;
            } else if (mat == 1) {
                beta_ws[idx] = e;
            } else {
                const float kv = prev_kappa[idx] + e;
                kappa_ws[idx]  = kv;
                kappa_out[idx] = kv;   // second output of the tuple
            }
        }
    }
}

// ---------------------------------------------------------------------------
// Kernel B: per-batch window. Compute phi[s] into LDS, then stream c[b,:,:]
// (512 KB per block) with coalesced float4 loads and reduce over s.
// grid = B blocks, block = 512 threads (16 waves).
// ---------------------------------------------------------------------------
__global__ __launch_bounds__(512)
void window_kernel(const float* __restrict__ alpha,
                   const float* __restrict__ beta,
                   const float* __restrict__ kappa,
                   const float* __restrict__ c,     // [B, S, L]
                   float* __restrict__ w)           // [B, L]
{
    __shared__ float  s_a[Kn], s_b[Kn], s_k[Kn];
    __shared__ float  s_phi[Sn];          // 4 KB
    __shared__ float4 s_red[16 * 32];     // 8 KB: 16 groups x 128 floats

    const int b   = blockIdx.x;
    const int tid = threadIdx.x;          // 0..511

    if (tid < Kn) {
        s_a[tid] = alpha[(size_t)b * Kn + tid];
        s_b[tid] = beta [(size_t)b * Kn + tid];
        s_k[tid] = kappa[(size_t)b * Kn + tid];
    }
    __syncthreads();

    // phi[b, s] = sum_k alpha_k * exp(-beta_k * (kappa_k - s)^2)
    for (int s = tid; s < Sn; s += 512) {
        const float su = (float)s;
        float acc = 0.f;
        #pragma unroll
        for (int k = 0; k < Kn; ++k) {
            const float d = s_k[k] - su;
            acc = __builtin_fmaf(s_a[k], __expf(-s_b[k] * d * d), acc);
        }
        s_phi[s] = acc;
    }
    __syncthreads();

    // w[b, l] = sum_s phi[s] * c[b, s, l]; float4 over l, 16 s-groups of 64
    const int l32 = tid & 31;             // float4 slot: l = 4*l32 .. 4*l32+3
    const int g   = tid >> 5;             // 0..15 -> s range [g*64, g*64+64)
    const float4* cp = reinterpret_cast<const float4*>(c + (size_t)b * Sn * Ln);

    float4 acc = {0.f, 0.f, 0.f, 0.f};
    const int s0 = g * (Sn / 16);
    #pragma unroll 4
    for (int s = s0; s < s0 + (Sn / 16); ++s) {
        const float  p  = s_phi[s];
        const float4 cv = cp[(size_t)s * 32 + l32];   // b128, fully coalesced
        acc.x = __builtin_fmaf(p, cv.x, acc.x);
        acc.y = __builtin_fmaf(p, cv.y, acc.y);
        acc.z = __builtin_fmaf(p, cv.z, acc.z);
        acc.w = __builtin_fmaf(p, cv.w, acc.w);
    }
    s_red[g * 32 + l32] = acc;
    __syncthreads();

    if (tid < 32) {
        float4 tot = {0.f, 0.f, 0.f, 0.f};
        #pragma unroll
        for (int gg = 0; gg < 16; ++gg) {
            const float4 v = s_red[gg * 32 + tid];
            tot.x += v.x; tot.y += v.y; tot.z += v.z; tot.w += v.w;
        }
        reinterpret_cast<float4*>(w + (size_t)b * Ln)[tid] = tot;
    }
}

// ---------------------------------------------------------------------------
extern "C" void kernel_launch(void* const* d_in, const int* in_sizes, int n_in,
                              void* d_out, int out_size, void* d_ws, size_t ws_size,
                              hipStream_t stream) {
    const float* x          = (const float*)d_in[0];   // [B, D]
    const float* c          = (const float*)d_in[1];   // [B, S, L]
    const float* prev_kappa = (const float*)d_in[2];   // [B, K]
    const float* Wa         = (const float*)d_in[3];   // [D, K]
    const float* ba         = (const float*)d_in[4];   // [K]
    const float* Wb         = (const float*)d_in[5];
    const float* bb         = (const float*)d_in[6];
    const float* Wk         = (const float*)d_in[7];
    const float* bk         = (const float*)d_in[8];

    float* w_out     = (float*)d_out;                         // [B, L]
    float* kappa_out = (float*)d_out + (size_t)Bn * Ln;       // [B, K]

    float* alpha_ws = (float*)d_ws;                           // [B, K]
    float* beta_ws  = alpha_ws + (size_t)Bn * Kn;             // [B, K]
    float* kappa_ws = beta_ws  + (size_t)Bn * Kn;             // [B, K]

    dim3 gridA(Bn / 16, 2, 3);
    proj_wmma_kernel<<<gridA, 32, 0, stream>>>(
        x, Wa, Wb, Wk, ba, bb, bk, prev_kappa,
        alpha_ws, beta_ws, kappa_ws, kappa_out);

    window_kernel<<<Bn, 512, 0, stream>>>(alpha_ws, beta_ws, kappa_ws, c, w_out);
}